// MultiScaleGrid_3796751089944
// MI455X (gfx1250) — compile-verified
//
#include <hip/hip_runtime.h>
#include <hip/hip_bf16.h>

typedef __attribute__((ext_vector_type(2))) float v2f;
typedef __attribute__((ext_vector_type(8))) float v8f;

// out[dst, bn] = sum_{src,k} (C[dst,src]/4) * spikes[src,k,bn]
//   C[dst,src] = w0*(|d|==1) + w1*(|d|==2) + w2*(|d|==4),  d = dst-src
// Computed as chained V_WMMA_F32_16X16X4_F32 over the 32-row (src,k) dimension,
// 16 columns (bn) per wave-tile. Rows 8..15 are zero padding.
__global__ void multiscale_grid_wmma(const float* __restrict__ spikes,
                                     const float* __restrict__ sw,
                                     float* __restrict__ out,
                                     int BN /* = B*N columns */) {
    const int lane   = threadIdx.x & 31;
    const int wave   = (blockIdx.x * blockDim.x + threadIdx.x) >> 5;
    const int nWaves = (gridDim.x * blockDim.x) >> 5;

    const float w0 = sw[0], w1 = sw[1], w2 = sw[2];

    // ---- A operands: loop-invariant. One v2f per src chunk j (K rows 4j..4j+3).
    // A[m][k] lives in lane (m + 16*(k>=2)), vgpr (k&1).  All 4 K-entries of
    // chunk j equal 0.25*C[m][j]; pad rows m>=8 are zero.
    const int m = lane & 15;
    v2f a[8];
#pragma unroll
    for (int j = 0; j < 8; ++j) {
        int d = m - j; if (d < 0) d = -d;
        float c = (d == 1) ? w0 : (d == 2) ? w1 : (d == 4) ? w2 : 0.0f;
        c = (m < 8) ? 0.25f * c : 0.0f;
        a[j].x = c;
        a[j].y = c;
    }

    // B 4x16 operand layout (mirror of A): lane l, vgpr r -> K = r + 2*(l>=16),
    // N = l & 15.
    const int col      = lane & 15;
    const int krow_off = (lane >> 4) << 1;   // 0 or 2

    const int nTiles = BN >> 4;              // 16 columns per tile
    for (int t = wave; t < nTiles; t += nWaves) {
        const int col0 = t << 4;
        const float* colp = spikes + col0 + col;

        // Prefetch next tile's column (lowers to global_prefetch_b8).
        if (t + nWaves < nTiles)
            __builtin_prefetch(spikes + ((t + nWaves) << 4) + col, 0, 3);

        v8f acc = {0.f, 0.f, 0.f, 0.f, 0.f, 0.f, 0.f, 0.f};
#pragma unroll
        for (int j = 0; j < 8; ++j) {
            const float* rp = colp + (size_t)(4 * j + krow_off) * (size_t)BN;
            v2f b;
            b.x = rp[0];
            b.y = rp[BN];
            // (neg_a, A, neg_b, B, c_mod, C, reuse_a, reuse_b)
            acc = __builtin_amdgcn_wmma_f32_16x16x4_f32(
                false, a[j], false, b, (short)0, acc, false, false);
        }

        // D rows 0..7 live in vgprs 0..7 of lanes 0..15.
        if (lane < 16) {
            float* op = out + col0 + lane;
#pragma unroll
            for (int r = 0; r < 8; ++r)
                op[(size_t)r * (size_t)BN] = acc[r];
        }
    }
}

extern "C" void kernel_launch(void* const* d_in, const int* in_sizes, int n_in,
                              void* d_out, int out_size, void* d_ws, size_t ws_size,
                              hipStream_t stream) {
    const float* spikes = (const float*)d_in[0];   // [M=8, K=4, B=32, N=16384] f32
    const float* sw     = (const float*)d_in[1];   // [3] f32
    float* out          = (float*)d_out;           // [8, 32, 16384] f32

    const int MK = 32;                       // M*K combined row count
    const int BN = in_sizes[0] / MK;         // 524288 columns

    const int threads = 256;                 // 8 wave32 per block
    const int blocks  = 512;                 // 4096 waves; ~8 tiles each
    multiscale_grid_wmma<<<blocks, threads, 0, stream>>>(spikes, sw, out, BN);
}